// myModel_61435212202090
// MI455X (gfx1250) — compile-verified
//
#include <hip/hip_runtime.h>
#include <hip/hip_bf16.h>

// Problem constants (match reference).
#define U_ 200000
#define I_ 100000
#define B_ 3
#define NNZ_ 1000000
#define D_ 64
#define L_ 2

typedef __attribute__((ext_vector_type(2))) float v2f;
typedef __attribute__((ext_vector_type(8))) float v8f;

// D = A(16x4 f32) * B(4x16 f32) + C(16x16 f32), wave32. Codegen-confirmed.
__device__ __forceinline__ v8f wmma4(v2f a, v2f b, v8f c) {
    return __builtin_amdgcn_wmma_f32_16x16x4_f32(
        /*neg_a=*/false, a, /*neg_b=*/false, b,
        /*c_mod=*/(short)0, c, /*reuse_a=*/false, /*reuse_b=*/false);
}

// ---------------------------------------------------------------------------
// Kernel 0: pack a row-major [K,64] weight into B-fragment order:
//   Wp[pr*64 + n] = (W[2*pr][n], W[2*pr+1][n])     pr = 0 .. K/2-1
// so a lane's B-fragment (rows kb,kb+1 at column n) is one global_load_b64.
// ---------------------------------------------------------------------------
__global__ void pack_w_kernel(const float* __restrict__ W, v2f* __restrict__ Wp,
                              int total /* = (K/2)*64 */) {
    int t = blockIdx.x * blockDim.x + threadIdx.x;
    if (t >= total) return;
    int n = t & 63;
    int k = (t >> 6) * 2;
    v2f v;
    v.x = W[k * 64 + n];
    v.y = W[(k + 1) * 64 + n];
    Wp[t] = v;
}

// ---------------------------------------------------------------------------
// Kernel 1: zero the SpMM accumulators in workspace.
// ---------------------------------------------------------------------------
__global__ void zero_ws_kernel(float4* __restrict__ p, long long n4) {
    long long t = (long long)blockIdx.x * blockDim.x + threadIdx.x;
    if (t < n4) p[t] = make_float4(0.f, 0.f, 0.f, 0.f);
}

// ---------------------------------------------------------------------------
// Kernel 2: edge-parallel multi-behavior SpMM scatter.
// One thread = one (edge, 4-float chunk of D). Does both directions:
//   ue_b[b, row, :] += v * item_emb[col, :]
//   ie_b[b, col, :] += v * user_emb[row, :]
// fp32 global atomics; accumulators are mostly L2-resident (230MB vs 192MB).
// ---------------------------------------------------------------------------
__global__ void spmm_scatter_kernel(const float* __restrict__ user_emb,
                                    const float* __restrict__ item_emb,
                                    const int* __restrict__ rows,
                                    const int* __restrict__ cols,
                                    const float* __restrict__ vals,
                                    float* __restrict__ ue_b,
                                    float* __restrict__ ie_b) {
    long long tid = (long long)blockIdx.x * blockDim.x + threadIdx.x;
    const long long total = (long long)B_ * NNZ_ * (D_ / 4);
    if (tid >= total) return;
    int c4 = (int)(tid & 15);          // which float4 chunk of D=64
    long long e = tid >> 4;            // flat edge index b*NNZ + n
    int b = (int)(e / NNZ_);
    int row = rows[e];
    int col = cols[e];
    float v = vals[e];

    const float4 it = *(const float4*)(item_emb + (long long)col * D_ + c4 * 4);
    float* up = ue_b + (((long long)b * U_ + row) * D_ + c4 * 4);
    atomicAdd(up + 0, v * it.x);
    atomicAdd(up + 1, v * it.y);
    atomicAdd(up + 2, v * it.z);
    atomicAdd(up + 3, v * it.w);

    const float4 us = *(const float4*)(user_emb + (long long)row * D_ + c4 * 4);
    float* ip = ie_b + (((long long)b * I_ + col) * D_ + c4 * 4);
    atomicAdd(ip + 0, v * us.x);
    atomicAdd(ip + 1, v * us.y);
    atomicAdd(ip + 2, v * us.z);
    atomicAdd(ip + 3, v * us.w);
}

// ---------------------------------------------------------------------------
// Kernel 3: fused dense tail, one wave per (16-row tile, variant).
//   variant 0..2 : X = Xb[variant]        -> out_b[variant]
//   variant 3    : X = mean_b Xb[b]       -> out_mean
// Per variant: for l in {0,1}: H_l = sigmoid(X @ W[l]) (WMMA f32 16x16x4),
// transpose H through LDS (C-layout -> A-layout), O += H_l @ Cw[l*64:,:].
// Weights come pre-packed in fragment order: one b64 load per WMMA.
// ---------------------------------------------------------------------------
__global__ void __launch_bounds__(128)
dense_tail_kernel(const float* __restrict__ Xb,   // [B, N, 64] accumulators
                  const v2f* __restrict__ Wp,     // packed [L*64, 64] layer weights
                  const v2f* __restrict__ Cwp,    // packed [128, 64] concat weights
                  float* __restrict__ out_mean,   // [N, 64]
                  float* __restrict__ out_b,      // [B, N, 64]
                  int N) {
    __shared__ float hstage[4 * 16 * 64];         // per-wave 16x64 staging tile

    const int lane = threadIdx.x & 31;
    const int wv   = threadIdx.x >> 5;            // wave id == variant (3 = mean)
    const int lo   = lane & 15;
    const int hi   = lane >> 4;
    const long long row0 = (long long)blockIdx.x * 16;
    float* hs = hstage + wv * (16 * 64);

    // ---- Load A-fragments of X (16x64) into registers.
    // A 16x4 layout: lane holds row M=lo; float2 at cols (4k + 2*hi, 4k + 2*hi + 1).
    v2f af[16];
    if (wv < 3) {
        const float* xp = Xb + (((long long)wv * N) + row0 + lo) * D_ + 2 * hi;
#pragma unroll
        for (int k = 0; k < 16; ++k) af[k] = *(const v2f*)(xp + 4 * k);
    } else {
#pragma unroll
        for (int k = 0; k < 16; ++k) { af[k].x = 0.f; af[k].y = 0.f; }
        for (int b = 0; b < B_; ++b) {
            const float* xp = Xb + (((long long)b * N) + row0 + lo) * D_ + 2 * hi;
#pragma unroll
            for (int k = 0; k < 16; ++k) af[k] += *(const v2f*)(xp + 4 * k);
        }
        const float inv = 1.0f / (float)B_;
#pragma unroll
        for (int k = 0; k < 16; ++k) { af[k].x *= inv; af[k].y *= inv; }
    }

    const v8f vzero = {0.f, 0.f, 0.f, 0.f, 0.f, 0.f, 0.f, 0.f};
    v8f o[4];
#pragma unroll
    for (int nt = 0; nt < 4; ++nt) o[nt] = vzero;

    for (int l = 0; l < L_; ++l) {
        // Packed row index for this lane: pr = l*32 + 2*ks + hi  (rows kb, kb+1).
        const v2f* WpL  = Wp  + (long long)(l * 32) * D_;
        const v2f* CwpL = Cwp + (long long)(l * 32) * D_;

        // ---- H = X @ W[l]  (16 K-steps x 4 N-tiles)
        v8f h[4];
#pragma unroll
        for (int nt = 0; nt < 4; ++nt) h[nt] = vzero;
#pragma unroll
        for (int ks = 0; ks < 16; ++ks) {
            const v2f a = af[ks];
            const v2f* wrow = WpL + (2 * ks + hi) * D_;
#pragma unroll
            for (int nt = 0; nt < 4; ++nt) {
                h[nt] = wmma4(a, wrow[nt * 16 + lo], h[nt]);
            }
        }

        // ---- sigmoid + stage to LDS (C-layout: row = r + 8*hi, col = nt*16 + lo)
#pragma unroll
        for (int nt = 0; nt < 4; ++nt) {
#pragma unroll
            for (int r = 0; r < 8; ++r) {
                const float s = 1.0f / (1.0f + __expf(-h[nt][r]));
                hs[(r + 8 * hi) * D_ + nt * 16 + lo] = s;
            }
        }
        // Cross-lane visibility of the staged tile within this wave.
        asm volatile("s_wait_dscnt 0" ::: "memory");

        // ---- O += H_l @ Cw[l*64 : (l+1)*64, :]
#pragma unroll
        for (int ks = 0; ks < 16; ++ks) {
            const int kb = ks * 4 + 2 * hi;
            const v2f a = *(const v2f*)(hs + lo * D_ + kb);   // ds_load_b64
            const v2f* crow = CwpL + (2 * ks + hi) * D_;
#pragma unroll
            for (int nt = 0; nt < 4; ++nt) {
                o[nt] = wmma4(a, crow[nt * 16 + lo], o[nt]);
            }
        }
        // Layer 1's LDS stores are same-wave in-order vs layer 0's loads -> no WAR fence.
    }

    // ---- Store O (C-layout scatter to row-major output).
    float* outp = (wv < 3)
        ? (out_b + (((long long)wv * N) + row0) * D_)
        : (out_mean + row0 * D_);
#pragma unroll
    for (int nt = 0; nt < 4; ++nt) {
#pragma unroll
        for (int r = 0; r < 8; ++r) {
            outp[(long long)(r + 8 * hi) * D_ + nt * 16 + lo] = o[nt][r];
        }
    }
}

// ---------------------------------------------------------------------------
extern "C" void kernel_launch(void* const* d_in, const int* in_sizes, int n_in,
                              void* d_out, int out_size, void* d_ws, size_t ws_size,
                              hipStream_t stream) {
    const float* user_emb = (const float*)d_in[0];
    const float* item_emb = (const float*)d_in[1];
    const int*   rows     = (const int*)d_in[2];
    const int*   cols     = (const int*)d_in[3];
    const float* vals     = (const float*)d_in[4];
    const float* u_w      = (const float*)d_in[5];
    const float* i_w      = (const float*)d_in[6];
    const float* u_cat_w  = (const float*)d_in[7];
    const float* i_cat_w  = (const float*)d_in[8];
    float* out = (float*)d_out;

    // Workspace layout: ue_b [B,U,64], ie_b [B,I,64] (fp32, 230.4 MB),
    // then 4 packed weight matrices (4 * 4096 v2f = 128 KB).
    float* ue_b = (float*)d_ws;
    float* ie_b = ue_b + (long long)B_ * U_ * D_;
    v2f* u_wp  = (v2f*)(ie_b + (long long)B_ * I_ * D_);
    v2f* i_wp  = u_wp  + 4096;   // (128/2)*64 fragments each
    v2f* u_cwp = i_wp  + 4096;
    v2f* i_cwp = u_cwp + 4096;

    // 0) pack weights into fragment order (tiny)
    pack_w_kernel<<<16, 256, 0, stream>>>(u_w,     u_wp,  4096);
    pack_w_kernel<<<16, 256, 0, stream>>>(i_w,     i_wp,  4096);
    pack_w_kernel<<<16, 256, 0, stream>>>(u_cat_w, u_cwp, 4096);
    pack_w_kernel<<<16, 256, 0, stream>>>(i_cat_w, i_cwp, 4096);

    // 1) zero accumulators
    const long long n_ws_floats = (long long)B_ * (U_ + I_) * D_;   // 57.6M
    const long long n4 = n_ws_floats / 4;
    zero_ws_kernel<<<(unsigned)((n4 + 255) / 256), 256, 0, stream>>>((float4*)d_ws, n4);

    // 2) SpMM scatter (both directions)
    const long long n_scatter = (long long)B_ * NNZ_ * (D_ / 4);     // 48M threads
    spmm_scatter_kernel<<<(unsigned)((n_scatter + 255) / 256), 256, 0, stream>>>(
        user_emb, item_emb, rows, cols, vals, ue_b, ie_b);

    // Output layout (flat, in return order):
    float* out_user_mean = out;                                      // [U,64]
    float* out_item_mean = out + (long long)U_ * D_;                 // [I,64]
    float* out_user_b    = out + (long long)(U_ + I_) * D_;          // [B,U,64]
    float* out_item_b    = out_user_b + (long long)B_ * U_ * D_;     // [B,I,64]

    // 3) fused dense tails (WMMA f32): user then item
    dense_tail_kernel<<<U_ / 16, 128, 0, stream>>>(ue_b, u_wp, u_cwp,
                                                   out_user_mean, out_user_b, U_);
    dense_tail_kernel<<<I_ / 16, 128, 0, stream>>>(ie_b, i_wp, i_cwp,
                                                   out_item_mean, out_item_b, I_);
}